// SIN_SLDS_VAE_37271726195415
// MI455X (gfx1250) — compile-verified
//
#include <hip/hip_runtime.h>
#include <hip/hip_bf16.h>
#include <math.h>

// ---------------- problem constants ----------------
#define TLEN 8192
#define NSTEP (TLEN - 1)
#define DD 32
#define KKMIX 16
#define LOG2PI_F 1.8378770664093453f

typedef __attribute__((ext_vector_type(2))) float v2f;
typedef __attribute__((ext_vector_type(8))) float v8f;

// ---------------- workspace layout (float offsets) ----------------
#define OFF_TAU      0
#define OFF_LAM      1024
#define OFF_ATLAM    2048
#define OFF_ATLAMA   3072
#define OFF_LAMA     4096
#define OFF_AMAT     5120
#define OFF_LAMB     6144
#define OFF_ATLAMB   6176
#define OFF_TAUMU    6208
#define OFF_BVEC     6240
#define OFF_TPM      6272
#define OFF_MLAST    6304
#define OFF_SCAL     6336
#define   SC_BTLAMB     0
#define   SC_LOGDET_LAM 1
#define   SC_LOGC0      2
#define   SC_LOGDET_TAU 3
#define   SC_MIXINIT    4
#define   SC_HMM_KL     5
#define   SC_HMM_LOGZ   6
#define   SC_LOGZQ      7
#define   SC_ACC0       8   // 10 accumulators from the smoother
#define OFF_KCONST   6400
#define OFF_CL       6416
#define OFF_COVT     7440
#define OFF_LAMAK    8464          // 16*1024
#define OFF_ATLAMAK  24848         // 16*1024
#define OFF_LAMBK    41232         // 16*32
#define OFF_ATLAMBK  41744         // 16*32
#define OFF_ALPHAS   42256         // 8191*16
#define OFF_JF       173312ull     // 8192*1024
#define OFF_HF       8561920ull    // 8192*32

// ---------------- output layout (float offsets) ----------------
#define O_Z      0ull
#define O_EX     262144ull
#define O_EXXT   524288ull
#define O_CROSS  8912896ull
#define O_CATES  17300480ull
#define O_KL     17431536ull

// =====================================================================
// helpers
// =====================================================================
__device__ __forceinline__ float mel(const float* M, int r, int c, bool tr) {
  return tr ? M[c * 32 + r] : M[r * 32 + c];
}

// Block (128 threads = 4 waves). Waves 0..3 compute the four 16x16 tiles of
//   Dst = base + sign * op(A) @ op(B)    (all 32x32, row-major, LDS)
// using V_WMMA_F32_16X16X4_F32 (f32 tensor path; 8 WMMA per tile).
__device__ void block_mm32(const float* A, bool tA, const float* B, bool tB,
                           const float* base, float sign, float* Dst) {
  __syncthreads();
  const int tid  = threadIdx.x;
  const int wave = tid >> 5;
  const int lane = tid & 31;
  if (wave < 4) {
    const int ti   = (wave >> 1) & 1;
    const int tj   = wave & 1;
    const int half = lane >> 4;   // lanes 0-15 vs 16-31
    const int l16  = lane & 15;
    v8f acc = {0.f, 0.f, 0.f, 0.f, 0.f, 0.f, 0.f, 0.f};
#pragma unroll
    for (int kk = 0; kk < 8; ++kk) {
      const int k0 = kk * 4;
      v2f a, b;
      // A frag 16x4: lanes 0-15 -> K = k0,k0+1 ; lanes 16-31 -> K = k0+2,k0+3
      const int arow = ti * 16 + l16;
      a.x = mel(A, arow, k0 + 2 * half + 0, tA);
      a.y = mel(A, arow, k0 + 2 * half + 1, tA);
      // B frag 4x16: VGPR0 -> K = k0 (+half), VGPR1 -> K = k0+2 (+half)
      const int bcol = tj * 16 + l16;
      b.x = mel(B, k0 + half,     bcol, tB);
      b.y = mel(B, k0 + 2 + half, bcol, tB);
      acc = __builtin_amdgcn_wmma_f32_16x16x4_f32(false, a, false, b,
                                                  (short)0, acc, false, false);
    }
#pragma unroll
    for (int v = 0; v < 8; ++v) {
      const int m = ti * 16 + v + half * 8;   // C/D layout: lanes0-15 M=v, lanes16-31 M=v+8
      const int n = tj * 16 + l16;
      const float b0 = base ? base[m * 32 + n] : 0.f;
      Dst[m * 32 + n] = b0 + sign * acc[v];
    }
  }
  __syncthreads();
}

// In-place lower Cholesky of a 32x32 SPD matrix in LDS (block cooperative).
__device__ void block_cholesky32(float* L) {
  const int tid = threadIdx.x;
  for (int k = 0; k < 32; ++k) {
    __syncthreads();
    if (tid == 0) L[k * 32 + k] = sqrtf(fmaxf(L[k * 32 + k], 1e-30f));
    __syncthreads();
    if (tid > k && tid < 32) L[tid * 32 + k] /= L[k * 32 + k];
    __syncthreads();
    const int span = (31 - k) * 32;
    for (int idx = tid; idx < span; idx += blockDim.x) {
      const int i = k + 1 + (idx >> 5);
      const int j = idx & 31;
      if (j > k && j <= i) L[i * 32 + j] -= L[i * 32 + k] * L[j * 32 + k];
    }
  }
  __syncthreads();
  for (int idx = tid; idx < 1024; idx += blockDim.x) {
    const int i = idx >> 5, j = idx & 31;
    if (j > i) L[idx] = 0.f;
  }
  __syncthreads();
}

// X = (L L^T)^{-1}; columns are independent -> 32 lanes, no inner barriers.
__device__ void block_spdinv32(const float* L, float* X, float* Ztmp) {
  const int tid = threadIdx.x;
  __syncthreads();
  if (tid < 32) {
    const int c = tid;
    for (int i = 0; i < 32; ++i) {            // L Z = I
      float s = (i == c) ? 1.f : 0.f;
      for (int j = 0; j < i; ++j) s -= L[i * 32 + j] * Ztmp[j * 32 + c];
      Ztmp[i * 32 + c] = s / L[i * 32 + i];
    }
    for (int i = 31; i >= 0; --i) {           // L^T X = Z
      float s = Ztmp[i * 32 + c];
      for (int j = i + 1; j < 32; ++j) s -= L[j * 32 + i] * X[j * 32 + c];
      X[i * 32 + c] = s / L[i * 32 + i];
    }
  }
  __syncthreads();
}

__device__ __forceinline__ unsigned wang(unsigned s) {
  s = (s ^ 61u) ^ (s >> 16); s *= 9u; s ^= s >> 4; s *= 0x27d4eb2du; s ^= s >> 15;
  return s;
}
__device__ float rnorm(int seed, int t, int d) {
  unsigned h1 = wang(0x9e3779b9u * (unsigned)t + 0x85ebca6bu * (unsigned)d +
                     (unsigned)seed * 0xc2b2ae35u + 1u);
  unsigned h2 = wang(h1 ^ 0x27220a95u);
  float u1 = (float)(h1 & 0xFFFFFFu) * (1.0f / 16777216.0f) + 1e-7f;
  float u2 = (float)(h2 & 0xFFFFFFu) * (1.0f / 16777216.0f);
  return sqrtf(-2.f * logf(u1)) * cosf(6.28318530717958f * u2);
}

// =====================================================================
// kernel 1: model constants (tau, lam, A, AT*lam*A, per-k mixture consts, ...)
// =====================================================================
__global__ void __launch_bounds__(128)
k_const(const float* Tau_p, const float* Lambda_p, const float* loc,
        const float* Xin, const float* prior_lam_g, const float* prior_Xg,
        const float* prior_tau, const float* prior_mu, float* W) {
  __shared__ float sA[1024], sB[1024], sC[1024], sL[1024];
  __shared__ float v1[32], v2[32];
  const int tid = threadIdx.x;

  // ---- tau = Tau_p Tau_p^T + 1e-8 I ----
  for (int e = tid; e < 1024; e += 128) sA[e] = Tau_p[e];
  block_mm32(sA, false, sA, true, nullptr, 1.f, sC);
  for (int e = tid; e < 1024; e += 128) {
    const int i = e >> 5, j = e & 31;
    float v = sC[e] + ((i == j) ? 1e-8f : 0.f);
    sC[e] = v; sL[e] = v; W[OFF_TAU + e] = v;
  }
  block_cholesky32(sL);
  if (tid == 0) { float s = 0; for (int k = 0; k < 32; ++k) s += logf(sL[k * 33]);
                  W[OFF_SCAL + SC_LOGDET_TAU] = 2.f * s; }
  __syncthreads();
  if (tid < 32) { float s = 0; for (int c = 0; c < 32; ++c) s += sC[tid * 32 + c] * loc[c];
                  v1[tid] = s; W[OFF_TAUMU + tid] = s; }
  __syncthreads();
  if (tid == 0) {
    float lt = 0; for (int c = 0; c < 32; ++c) lt += loc[c] * v1[c];
    W[OFF_SCAL + SC_LOGC0] = -0.5f * lt + 0.5f * W[OFF_SCAL + SC_LOGDET_TAU]
                             - 0.5f * 32.f * LOG2PI_F;
  }
  __syncthreads();

  // ---- lam = Lambda_p Lambda_p^T + 1e-8 I ----
  for (int e = tid; e < 1024; e += 128) sA[e] = Lambda_p[e];
  block_mm32(sA, false, sA, true, nullptr, 1.f, sC);
  for (int e = tid; e < 1024; e += 128) {
    const int i = e >> 5, j = e & 31;
    float v = sC[e] + ((i == j) ? 1e-8f : 0.f);
    sB[e] = v; sL[e] = v; W[OFF_LAM + e] = v;
  }
  block_cholesky32(sL);
  if (tid == 0) { float s = 0; for (int k = 0; k < 32; ++k) s += logf(sL[k * 33]);
                  W[OFF_SCAL + SC_LOGDET_LAM] = 2.f * s; }
  __syncthreads();

  // ---- A, b from X (D x D+1) ----
  for (int e = tid; e < 1024; e += 128) {
    const int i = e >> 5, j = e & 31;
    sA[e] = Xin[i * 33 + j]; W[OFF_AMAT + e] = sA[e];
  }
  if (tid < 32) { v2[tid] = Xin[tid * 33 + 32]; W[OFF_BVEC + tid] = v2[tid]; }
  block_mm32(sA, true, sB, false, nullptr, 1.f, sC);      // ATlam = A^T lam
  for (int e = tid; e < 1024; e += 128) W[OFF_ATLAM + e] = sC[e];
  block_mm32(sC, false, sA, false, nullptr, 1.f, sL);     // ATlamA
  for (int e = tid; e < 1024; e += 128) W[OFF_ATLAMA + e] = sL[e];
  block_mm32(sB, false, sA, false, nullptr, 1.f, sL);     // lamA
  for (int e = tid; e < 1024; e += 128) W[OFF_LAMA + e] = sL[e];
  __syncthreads();
  if (tid < 32) { float s = 0; for (int c = 0; c < 32; ++c) s += sB[tid * 32 + c] * v2[c];
                  v1[tid] = s; W[OFF_LAMB + tid] = s; }
  __syncthreads();
  if (tid < 32) { float s = 0; for (int i = 0; i < 32; ++i) s += sA[i * 32 + tid] * v1[i];
                  W[OFF_ATLAMB + tid] = s; }
  if (tid == 0) { float s = 0; for (int c = 0; c < 32; ++c) s += v2[c] * v1[c];
                  W[OFF_SCAL + SC_BTLAMB] = s; }
  __syncthreads();

  // ---- per-k mixture constants ----
  for (int k = 0; k < KKMIX; ++k) {
    for (int e = tid; e < 1024; e += 128) {
      const int i = e >> 5, j = e & 31;
      sB[e] = prior_lam_g[k * 1024 + e];
      sA[e] = prior_Xg[k * 1056 + i * 33 + j];
    }
    if (tid < 32) v2[tid] = prior_Xg[k * 1056 + tid * 33 + 32];
    block_mm32(sB, false, sA, false, nullptr, 1.f, sC);   // lamAk
    for (int e = tid; e < 1024; e += 128) W[OFF_LAMAK + k * 1024 + e] = sC[e];
    block_mm32(sA, true, sC, false, nullptr, 1.f, sL);    // ATlamAk
    for (int e = tid; e < 1024; e += 128) W[OFF_ATLAMAK + k * 1024 + e] = sL[e];
    __syncthreads();
    if (tid < 32) { float s = 0; for (int c = 0; c < 32; ++c) s += sB[tid * 32 + c] * v2[c];
                    v1[tid] = s; W[OFF_LAMBK + k * 32 + tid] = s; }
    __syncthreads();
    if (tid < 32) { float s = 0; for (int i = 0; i < 32; ++i) s += sA[i * 32 + tid] * v1[i];
                    W[OFF_ATLAMBK + k * 32 + tid] = s; }
    for (int e = tid; e < 1024; e += 128) sL[e] = sB[e];
    block_cholesky32(sL);
    if (tid == 0) {
      float s = 0; for (int kk = 0; kk < 32; ++kk) s += logf(sL[kk * 33]);
      float bt = 0; for (int c = 0; c < 32; ++c) bt += v2[c] * v1[c];
      W[OFF_KCONST + k] = -0.5f * bt + 0.5f * (2.f * s);
    }
    __syncthreads();
  }

  // ---- prior_tau / prior_mu init constants ----
  for (int e = tid; e < 1024; e += 128) { sB[e] = prior_tau[e]; sL[e] = prior_tau[e]; }
  __syncthreads();
  if (tid < 32) { float s = 0; for (int c = 0; c < 32; ++c) s += sB[tid * 32 + c] * prior_mu[c];
                  v1[tid] = s; W[OFF_TPM + tid] = s; }
  block_cholesky32(sL);
  if (tid == 0) {
    float s = 0; for (int k = 0; k < 32; ++k) s += logf(sL[k * 33]);
    float mt = 0; for (int c = 0; c < 32; ++c) mt += prior_mu[c] * v1[c];
    W[OFF_SCAL + SC_MIXINIT] = -0.5f * mt + s - 0.5f * 32.f * LOG2PI_F;
  }
}

// =====================================================================
// kernel 2: HMM forward/backward (emissions are zero) -> cat_es, hmm_kl
// =====================================================================
__global__ void __launch_bounds__(64)
k_hmm(const float* init_lps, const float* trans_lps,
      const float* init_lps_prior, const float* trans_lps_prior,
      float* W, float* out) {
  __shared__ float Tr[256], a[16], a2[16], beta[16], nb[16], EZ[256];
  __shared__ float sLogZ;
  const int tid = threadIdx.x;
  for (int e = tid; e < 256; e += 64) { Tr[e] = trans_lps[e]; EZ[e] = 0.f; }
  if (tid < 16) a[tid] = init_lps[tid];
  __syncthreads();
  float* alph = W + OFF_ALPHAS;
  if (tid < 16) alph[tid] = a[tid];
  __syncthreads();
  for (int n = 1; n < NSTEP; ++n) {
    if (tid < 16) {
      const int j = tid;
      float m = -1e30f;
      for (int i = 0; i < 16; ++i) m = fmaxf(m, a[i] + Tr[i * 16 + j]);
      float s = 0.f;
      for (int i = 0; i < 16; ++i) s += expf(a[i] + Tr[i * 16 + j] - m);
      a2[j] = m + logf(s);
    }
    __syncthreads();
    if (tid < 16) { a[tid] = a2[tid]; alph[(size_t)n * 16 + tid] = a2[tid]; }
    __syncthreads();
  }
  if (tid == 0) {
    float m = -1e30f; for (int i = 0; i < 16; ++i) m = fmaxf(m, a[i]);
    float s = 0.f;    for (int i = 0; i < 16; ++i) s += expf(a[i] - m);
    sLogZ = m + logf(s);
    W[OFF_SCAL + SC_HMM_LOGZ] = sLogZ;
  }
  __syncthreads();
  const float logZ = sLogZ;
  if (tid < 16) {
    beta[tid] = 0.f;
    out[O_CATES + (size_t)(NSTEP - 1) * 16 + tid] =
        expf(alph[(size_t)(NSTEP - 1) * 16 + tid] - logZ);
  }
  __syncthreads();
  for (int n = NSTEP - 2; n >= 0; --n) {
    for (int e = tid; e < 256; e += 64) {
      const int i = e >> 4, j = e & 15;
      EZ[e] += expf(alph[(size_t)n * 16 + i] + Tr[e] + beta[j] - logZ);
    }
    if (tid < 16) {
      const int i = tid;
      float m = -1e30f;
      for (int j = 0; j < 16; ++j) m = fmaxf(m, Tr[i * 16 + j] + beta[j]);
      float s = 0.f;
      for (int j = 0; j < 16; ++j) s += expf(Tr[i * 16 + j] + beta[j] - m);
      nb[i] = m + logf(s);
    }
    __syncthreads();
    if (tid < 16) {
      beta[tid] = nb[tid];
      out[O_CATES + (size_t)n * 16 + tid] = expf(alph[(size_t)n * 16 + tid] + beta[tid] - logZ);
    }
    __syncthreads();
  }
  if (tid == 0) {
    double kl = 0.0;
    for (int j = 0; j < 16; ++j)
      kl += (double)expf(alph[j] + beta[j] - logZ) * (double)(init_lps[j] - init_lps_prior[j]);
    for (int e = 0; e < 256; ++e)
      kl += (double)EZ[e] * (double)(trans_lps[e] - trans_lps_prior[e]);
    kl -= (double)logZ;
    W[OFF_SCAL + SC_HMM_KL] = (float)kl;
  }
}

// =====================================================================
// kernel 3: forward information filter (serial scan, WMMA GEMMs per step)
// =====================================================================
__global__ void __launch_bounds__(128)
k_filt(const float* recog_J, const float* recog_h, float* W, float* out) {
  __shared__ float cLam[1024], cATlam[1024], cATlamA[1024], cLamA[1024];
  __shared__ float sJ[1024], sL[1024], sPinv[1024], sY[1024], sZ[1024];
  __shared__ float sh[32], su[32], smu[32], cLamb[32], cATlamb[32];
  __shared__ float sScal[2];   // [0]=lc, [1]=logdetM
  const int tid = threadIdx.x;
  float* Jf = W + OFF_JF;
  float* hf = W + OFF_HF;
  for (int e = tid; e < 1024; e += 128) {
    cLam[e] = W[OFF_LAM + e]; cATlam[e] = W[OFF_ATLAM + e];
    cATlamA[e] = W[OFF_ATLAMA + e]; cLamA[e] = W[OFF_LAMA + e];
    float j0 = W[OFF_TAU + e] + recog_J[e];
    sJ[e] = j0; Jf[e] = j0;
  }
  if (tid < 32) {
    cLamb[tid] = W[OFF_LAMB + tid]; cATlamb[tid] = W[OFF_ATLAMB + tid];
    float h0 = W[OFF_TAUMU + tid] + recog_h[tid];
    sh[tid] = h0; hf[tid] = h0;
  }
  if (tid == 0) sScal[0] = W[OFF_SCAL + SC_LOGC0];
  const float bTlamb = W[OFF_SCAL + SC_BTLAMB];
  const float ldlam  = W[OFF_SCAL + SC_LOGDET_LAM];
  __syncthreads();

  for (int t = 1; t < TLEN; ++t) {
    const float* RJ = recog_J + (size_t)t * 1024;
    if (t + 1 < TLEN)
      __builtin_prefetch(recog_J + (size_t)(t + 1) * 1024 + tid * 8, 0, 1);
    for (int e = tid; e < 1024; e += 128) {
      const int i = e >> 5, j = e & 31;
      sL[e] = sJ[e] + cATlamA[e] + ((i == j) ? 1e-6f : 0.f);
    }
    block_cholesky32(sL);
    if (tid == 0) { float s = 0; for (int k = 0; k < 32; ++k) s += logf(sL[k * 33]);
                    sScal[1] = 2.f * s; }
    block_spdinv32(sL, sPinv, sY);
    block_mm32(sPinv, false, cATlam, false, nullptr, 1.f, sY);   // Y = M^-1 ATlam
    block_mm32(cLamA, false, sY, false, nullptr, 1.f, sZ);       // Z = lamA Y
    if (tid < 32) su[tid] = sh[tid] - cATlamb[tid];
    __syncthreads();
    if (tid < 32) { float s = 0; for (int c = 0; c < 32; ++c) s += sPinv[tid * 32 + c] * su[c];
                    smu[tid] = s; }
    __syncthreads();
    if (tid == 0) {
      float um = 0; for (int c = 0; c < 32; ++c) um += su[c] * smu[c];
      sScal[0] += -0.5f * bTlamb + 0.5f * ldlam + 0.5f * um - 0.5f * sScal[1];
    }
    for (int e = tid; e < 1024; e += 128) {
      float v = cLam[e] + RJ[e] - sZ[e];
      sJ[e] = v; Jf[(size_t)t * 1024 + e] = v;
    }
    if (tid < 32) {
      float s = 0; for (int c = 0; c < 32; ++c) s += cLamA[tid * 32 + c] * smu[c];
      float v = cLamb[tid] + recog_h[(size_t)t * 32 + tid] + s;
      sh[tid] = v; hf[(size_t)t * 32 + tid] = v;
    }
    __syncthreads();
  }

  // terminal: Cl, mlast, CovT, logZq, Ex/ExxT at T-1
  for (int e = tid; e < 1024; e += 128) {
    const int i = e >> 5, j = e & 31;
    sL[e] = sJ[e] + ((i == j) ? 1e-6f : 0.f);
  }
  block_cholesky32(sL);
  block_spdinv32(sL, sPinv, sY);                     // CovT
  if (tid < 32) { float s = 0; for (int c = 0; c < 32; ++c) s += sPinv[tid * 32 + c] * sh[c];
                  smu[tid] = s; }                    // mlast
  __syncthreads();
  if (tid == 0) {
    float hm = 0; for (int c = 0; c < 32; ++c) hm += sh[c] * smu[c];
    float sld = 0; for (int k = 0; k < 32; ++k) sld += logf(sL[k * 33]);
    W[OFF_SCAL + SC_LOGZQ] = sScal[0] + 0.5f * hm + 0.5f * 32.f * LOG2PI_F - sld;
  }
  for (int e = tid; e < 1024; e += 128) {
    const int i = e >> 5, j = e & 31;
    W[OFF_CL + e] = sL[e]; W[OFF_COVT + e] = sPinv[e];
    out[O_EXXT + (size_t)(TLEN - 1) * 1024 + e] = sPinv[e] + smu[i] * smu[j];
  }
  if (tid < 32) { W[OFF_MLAST + tid] = smu[tid];
                  out[O_EX + (size_t)(TLEN - 1) * 32 + tid] = smu[tid]; }
}

// =====================================================================
// kernel 4: backward RTS smoother + fused KL trace accumulations
// =====================================================================
__global__ void __launch_bounds__(128)
k_smooth(const float* recog_J, const float* recog_h, const float* prior_lam_g,
         float* W, float* out) {
  __shared__ float cLam[1024], cATlam[1024], cATlamA[1024], cLamA[1024];
  __shared__ float sL[1024], sPinv[1024], sG[1024], sW1[1024], sS[1024],
                   sCross[1024], sCov[1024];
  __shared__ float smn[32], smnew[32], sg[32], shv[32], cATlamb[32], cLamb[32], swk[16];
  __shared__ double red[128];
  const int tid = threadIdx.x;
  const float* Jf = W + OFF_JF;
  const float* hf = W + OFF_HF;
  for (int e = tid; e < 1024; e += 128) {
    cLam[e] = W[OFF_LAM + e]; cATlam[e] = W[OFF_ATLAM + e];
    cATlamA[e] = W[OFF_ATLAMA + e]; cLamA[e] = W[OFF_LAMA + e];
    sCov[e] = W[OFF_COVT + e];
  }
  if (tid < 32) { smn[tid] = W[OFF_MLAST + tid];
                  cATlamb[tid] = W[OFF_ATLAMB + tid]; cLamb[tid] = W[OFF_LAMB + tid]; }
  __syncthreads();

  double a0 = 0, a1 = 0, a2 = 0, a3 = 0, a4 = 0, a5 = 0, a6 = 0, a7 = 0, a8 = 0, a9 = 0;

  for (int t = TLEN - 2; t >= 0; --t) {
    for (int e = tid; e < 1024; e += 128) {
      const int i = e >> 5, j = e & 31;
      sL[e] = Jf[(size_t)t * 1024 + e] + cATlamA[e] + ((i == j) ? 1e-6f : 0.f);
    }
    if (tid < 32) shv[tid] = hf[(size_t)t * 32 + tid];
    if (tid < 16) swk[tid] = out[O_CATES + (size_t)t * 16 + tid];
    block_cholesky32(sL);
    block_spdinv32(sL, sPinv, sW1);
    block_mm32(sPinv, false, cATlam, false, nullptr, 1.f, sG);   // G = P^-1 ATlam
    if (tid < 32) { float s = 0;
      for (int c = 0; c < 32; ++c) s += sPinv[tid * 32 + c] * (shv[c] - cATlamb[c]);
      sg[tid] = s; }
    __syncthreads();
    if (tid < 32) { float s = sg[tid];
      for (int c = 0; c < 32; ++c) s += sG[tid * 32 + c] * smn[c];
      smnew[tid] = s; }
    for (int e = tid; e < 1024; e += 128) {
      const int i = e >> 5, j = e & 31;
      sS[e] = sCov[e] + smn[i] * smn[j];                         // = ExxT[t+1]
    }
    __syncthreads();
    block_mm32(sG, false, sCov, false, nullptr, 1.f, sW1);       // G Cov_n
    block_mm32(sW1, false, sG, true, sPinv, 1.f, sCov);          // Cov = P^-1 + G Cov_n G^T
    block_mm32(sG, false, sS, false, nullptr, 1.f, sCross);      // G (Cov_n + m m^T)
    for (int e = tid; e < 1024; e += 128) {
      const int i = e >> 5, j = e & 31;
      sCross[e] += sg[i] * smn[j];
    }
    __syncthreads();
    for (int e = tid; e < 1024; e += 128) {
      const int i = e >> 5, j = e & 31;
      const float crT = sCross[j * 32 + i];                      // Cross[t][i][j]
      const float ext = sCov[e] + smnew[i] * smnew[j];           // ExxT[t]
      out[O_EXXT + (size_t)t * 1024 + e]  = ext;
      out[O_CROSS + (size_t)t * 1024 + e] = crT;
      a0 += (double)recog_J[(size_t)t * 1024 + e] * (double)ext;
      a2 += (double)cLam[e] * (double)sS[e];
      a3 += (double)cLamA[e] * (double)crT;
      a4 += (double)cATlamA[e] * (double)ext;
      float mq = 0.f;
#pragma unroll 4
      for (int k = 0; k < KKMIX; ++k) {
        mq += swk[k] * (-0.5f * prior_lam_g[k * 1024 + e] * sS[e]
                        + W[OFF_LAMAK + k * 1024 + e] * crT
                        - 0.5f * W[OFF_ATLAMAK + k * 1024 + e] * ext);
      }
      a7 += (double)mq;
    }
    if (tid < 32) {
      out[O_EX + (size_t)t * 32 + tid] = smnew[tid];
      a1 += (double)recog_h[(size_t)t * 32 + tid] * (double)smnew[tid];
      a5 += (double)cLamb[tid] * (double)smn[tid];
      a6 += (double)cATlamb[tid] * (double)smnew[tid];
      float vq = 0.f;
      for (int k = 0; k < KKMIX; ++k)
        vq += swk[k] * (W[OFF_LAMBK + k * 32 + tid] * smn[tid]
                        - W[OFF_ATLAMBK + k * 32 + tid] * smnew[tid]);
      a8 += (double)vq;
    }
    if (tid == 0) { float c9 = 0;
      for (int k = 0; k < KKMIX; ++k) c9 += swk[k] * W[OFF_KCONST + k];
      a9 += (double)c9; }
    __syncthreads();
    if (tid < 32) smn[tid] = smnew[tid];
    __syncthreads();
  }

  double accs[10] = {a0, a1, a2, a3, a4, a5, a6, a7, a8, a9};
  for (int q = 0; q < 10; ++q) {
    red[tid] = accs[q];
    __syncthreads();
    for (int s = 64; s > 0; s >>= 1) { if (tid < s) red[tid] += red[tid + s]; __syncthreads(); }
    if (tid == 0) W[OFF_SCAL + SC_ACC0 + q] = (float)red[0];
    __syncthreads();
  }
}

// =====================================================================
// kernel 5: backward sampling pass
// =====================================================================
__global__ void __launch_bounds__(64)
k_samp(const int* seedp, float* W, float* out) {
  __shared__ float cATlam[1024], cATlamA[1024], sL[1024];
  __shared__ float xnext[32], rhs[32], cATlamb[32];
  const int tid = threadIdx.x;
  const int seed = seedp[0];
  const float* Jf = W + OFF_JF;
  const float* hf = W + OFF_HF;
  for (int e = tid; e < 1024; e += 64) {
    cATlam[e] = W[OFF_ATLAM + e]; cATlamA[e] = W[OFF_ATLAMA + e];
    sL[e] = W[OFF_CL + e];
  }
  if (tid < 32) cATlamb[tid] = W[OFF_ATLAMB + tid];
  __syncthreads();
  if (tid == 0) {
    float ev[32], y[32];
    for (int d = 0; d < 32; ++d) ev[d] = rnorm(seed, TLEN - 1, d);
    for (int i = 31; i >= 0; --i) {
      float s = ev[i];
      for (int j = i + 1; j < 32; ++j) s -= sL[j * 32 + i] * y[j];
      y[i] = s / sL[i * 33];
    }
    for (int d = 0; d < 32; ++d) {
      float z = W[OFF_MLAST + d] + y[d];
      xnext[d] = z; out[O_Z + (size_t)(TLEN - 1) * 32 + d] = z;
    }
  }
  __syncthreads();
  for (int t = TLEN - 2; t >= 0; --t) {
    for (int e = tid; e < 1024; e += 64) {
      const int i = e >> 5, j = e & 31;
      sL[e] = Jf[(size_t)t * 1024 + e] + cATlamA[e] + ((i == j) ? 1e-6f : 0.f);
    }
    block_cholesky32(sL);
    if (tid < 32) {
      float s = hf[(size_t)t * 32 + tid] - cATlamb[tid];
      for (int c = 0; c < 32; ++c) s += cATlam[tid * 32 + c] * xnext[c];
      rhs[tid] = s;
    }
    __syncthreads();
    if (tid == 0) {
      float aa[32], mean[32], y[32], ev[32];
      for (int i = 0; i < 32; ++i) {
        float s = rhs[i];
        for (int j = 0; j < i; ++j) s -= sL[i * 32 + j] * aa[j];
        aa[i] = s / sL[i * 33];
      }
      for (int i = 31; i >= 0; --i) {
        float s = aa[i];
        for (int j = i + 1; j < 32; ++j) s -= sL[j * 32 + i] * mean[j];
        mean[i] = s / sL[i * 33];
      }
      for (int d = 0; d < 32; ++d) ev[d] = rnorm(seed, t, d);
      for (int i = 31; i >= 0; --i) {
        float s = ev[i];
        for (int j = i + 1; j < 32; ++j) s -= sL[j * 32 + i] * y[j];
        y[i] = s / sL[i * 33];
      }
      for (int d = 0; d < 32; ++d) {
        float z = mean[d] + y[d];
        xnext[d] = z; out[O_Z + (size_t)t * 32 + d] = z;
      }
    }
    __syncthreads();
  }
}

// =====================================================================
// kernel 6: final KL assembly
// =====================================================================
__global__ void __launch_bounds__(128)
k_final(const float* recog_J, const float* recog_h, const float* loc,
        const float* prior_tau, float* W, float* out) {
  __shared__ double red[128];
  const int tid = threadIdx.x;
  double vals[7] = {0, 0, 0, 0, 0, 0, 0};
  const float* Exx0 = out + O_EXXT;
  const float* ExxL = out + O_EXXT + (size_t)(TLEN - 1) * 1024;
  for (int e = tid; e < 1024; e += 128) {
    vals[0] += (double)W[OFF_TAU + e] * (double)Exx0[e];
    vals[1] += (double)prior_tau[e] * (double)Exx0[e];
    vals[2] += (double)recog_J[(size_t)(TLEN - 1) * 1024 + e] * (double)ExxL[e];
  }
  if (tid < 32) {
    const float ex0 = out[O_EX + tid];
    const float exL = out[O_EX + (size_t)(TLEN - 1) * 32 + tid];
    vals[3] = (double)ex0 * (double)W[OFF_TAUMU + tid];
    vals[4] = (double)ex0 * (double)W[OFF_TPM + tid];
    vals[5] = (double)recog_h[(size_t)(TLEN - 1) * 32 + tid] * (double)exL;
    vals[6] = (double)loc[tid] * (double)W[OFF_TAUMU + tid];
  }
  double res[7];
  for (int q = 0; q < 7; ++q) {
    red[tid] = vals[q];
    __syncthreads();
    for (int s = 64; s > 0; s >>= 1) { if (tid < s) red[tid] += red[tid + s]; __syncthreads(); }
    res[q] = red[0];
    __syncthreads();
  }
  if (tid == 0) {
    const double Nn = (double)NSTEP, Dd = 32.0, L2P = (double)LOG2PI_F;
    double acc[10];
    for (int q = 0; q < 10; ++q) acc[q] = (double)W[OFF_SCAL + SC_ACC0 + q];
    const double ldtau = W[OFF_SCAL + SC_LOGDET_TAU];
    const double ldlam = W[OFF_SCAL + SC_LOGDET_LAM];
    const double bT    = W[OFF_SCAL + SC_BTLAMB];
    const double trans = -0.5 * acc[2] + acc[3] - 0.5 * acc[4] + acc[5] - acc[6]
                         + Nn * (-0.5 * bT + 0.5 * ldlam - 0.5 * Dd * L2P);
    const double init  = -0.5 * res[0] + res[3] - 0.5 * res[6] + 0.5 * ldtau - 0.5 * Dd * L2P;
    const double recog = -0.5 * (acc[0] + res[2]) + (acc[1] + res[5]);
    const double mix_t = acc[7] + acc[8] + acc[9] - 0.5 * Nn * Dd * L2P;
    const double mix_i = -0.5 * res[1] + res[4] + (double)W[OFF_SCAL + SC_MIXINIT];
    const double lds_kl = (init + trans) + recog
                          - (double)W[OFF_SCAL + SC_LOGZQ] - (mix_i + mix_t);
    out[O_KL] = (float)(lds_kl + (double)W[OFF_SCAL + SC_HMM_KL]);
  }
}

// =====================================================================
extern "C" void kernel_launch(void* const* d_in, const int* in_sizes, int n_in,
                              void* d_out, int out_size, void* d_ws, size_t ws_size,
                              hipStream_t stream) {
  const float* recog_J         = (const float*)d_in[0];
  const float* recog_h         = (const float*)d_in[1];
  const float* loc             = (const float*)d_in[2];
  const float* Tau_p           = (const float*)d_in[3];
  const float* Lambda_p        = (const float*)d_in[4];
  const float* X               = (const float*)d_in[5];
  const float* init_lps        = (const float*)d_in[6];
  const float* trans_lps       = (const float*)d_in[7];
  const float* prior_lam       = (const float*)d_in[8];
  const float* prior_X         = (const float*)d_in[9];
  const float* prior_tau       = (const float*)d_in[10];
  const float* prior_mu        = (const float*)d_in[11];
  const float* init_lps_prior  = (const float*)d_in[12];
  const float* trans_lps_prior = (const float*)d_in[13];
  const int*   seed            = (const int*)d_in[14];
  float* W   = (float*)d_ws;
  float* out = (float*)d_out;

  k_const <<<1, 128, 0, stream>>>(Tau_p, Lambda_p, loc, X, prior_lam, prior_X,
                                  prior_tau, prior_mu, W);
  k_hmm   <<<1, 64, 0, stream>>>(init_lps, trans_lps, init_lps_prior, trans_lps_prior, W, out);
  k_filt  <<<1, 128, 0, stream>>>(recog_J, recog_h, W, out);
  k_smooth<<<1, 128, 0, stream>>>(recog_J, recog_h, prior_lam, W, out);
  k_samp  <<<1, 64, 0, stream>>>(seed, W, out);
  k_final <<<1, 128, 0, stream>>>(recog_J, recog_h, loc, prior_tau, W, out);
}